// LUTLayer_52072183496901
// MI455X (gfx1250) — compile-verified
//
#include <hip/hip_runtime.h>
#include <stdint.h>
#include <stddef.h>

typedef unsigned int u32;
typedef unsigned long long u64;

// ---- CDNA5 async global->LDS path (gfx1250), probe-guarded ------------------
#if defined(__HIP_DEVICE_COMPILE__) && defined(__gfx1250__)
  #if __has_builtin(__builtin_amdgcn_global_load_async_to_lds_b128) && \
      __has_builtin(__builtin_amdgcn_s_wait_asynccnt)
    #define USE_ASYNC_LDS 1
  #endif
#endif

#if defined(USE_ASYNC_LDS)
// Builtin prototype (from hipcc diagnostic): param0 = int4 AS1*, param1 = int4 AS3*.
typedef int v4i_ __attribute__((ext_vector_type(4)));
typedef __attribute__((address_space(1))) v4i_ g_v4i;
typedef __attribute__((address_space(3))) v4i_ l_v4i;
#endif

__device__ __forceinline__ u32 ballot32(bool p) {
#if defined(__HIP_DEVICE_COMPILE__) && __has_builtin(__builtin_amdgcn_ballot_w32)
  return __builtin_amdgcn_ballot_w32(p);
#else
  return (u32)__ballot(p);
#endif
}

// -----------------------------------------------------------------------------
// Kernel 1: bit-pack a float array (bit j of word w = pred(src[w*32+j])).
// strict_pos=0 -> pred is (v != 0)   (for x, values are exactly 0.0/1.0)
// strict_pos=1 -> pred is (v > 0)    (for the LUT sign mask)
// n must be a multiple of 32 (holds: 512*8192 and 4096*64).
// -----------------------------------------------------------------------------
__global__ __launch_bounds__(256) void pack_bits_kernel(
    const float* __restrict__ src, u32* __restrict__ dst, int n, int strict_pos) {
  int g = blockIdx.x * 256 + threadIdx.x;
  bool pred = false;
  if (g < n) {
    float v = src[g];
    pred = strict_pos ? (v > 0.0f) : (v != 0.0f);
  }
  u32 m = ballot32(pred);
  if ((threadIdx.x & 31u) == 0u && g < n) dst[g >> 5] = m;
}

// -----------------------------------------------------------------------------
// Kernel 2: main LUT forward.
//   block = 128 threads, covers BB=8 batch rows x OT=512 outputs.
//   Stage 1: async-copy (gfx1250 global_load_async_to_lds_b128) the 8 packed
//            x rows (8 KB) into LDS, drain with s_wait_asynccnt.
//   Stage 2: transpose in LDS to xb[w][b] with a padded stride of 12 words so
//            that one ds_load_b128 fetches word w of 4 batch rows at once
//            (48 B stride keeps 16 B alignment and spreads banks: gcd(12,64)=4).
//   Stage 3: per output: 6 word offsets -> 12 b128 LDS probes serve all 8 rows;
//            addr -> out = (lut_signmask >> addr) & 1; coalesced float4 stores.
// -----------------------------------------------------------------------------
#define WORDS 256   // I/32 = 8192/32
#define BB    8
#define OT    512
#define TSTR  12    // transposed stride (words) per w: 8 rows + 4 pad

__global__ __launch_bounds__(128) void lut_forward_kernel(
    const u32* __restrict__ xbits, const int* __restrict__ mapping,
    const u64* __restrict__ lutbits, float* __restrict__ out, int Onum) {
  __shared__ __align__(16) u32 xlin[BB * WORDS];    //  8 KB linear staging
  __shared__ __align__(16) u32 xb[WORDS * TSTR];    // 12 KB transposed

  const int t     = threadIdx.x;
  const int obase = blockIdx.x * OT;
  const int bbase = blockIdx.y * BB;
  const u32* src  = xbits + (size_t)bbase * WORDS;  // 2048 contiguous words

  // --- Stage 1: async global->LDS copy of 8 KB (128 thr * 16 B * 4 iters) ---
#if defined(USE_ASYNC_LDS)
  #pragma unroll
  for (int it = 0; it < 4; ++it) {
    int off = (it * 128 + t) * 4;  // word index, 16B granularity
    __builtin_amdgcn_global_load_async_to_lds_b128(
        (g_v4i*)(src + off), (l_v4i*)(xlin + off), 0, 0);
  }
  __builtin_amdgcn_s_wait_asynccnt(0);
#else
  #pragma unroll
  for (int it = 0; it < 4; ++it) {
    int wi = it * 128 + t;
    ((uint4*)xlin)[wi] = ((const uint4*)src)[wi];
  }
#endif
  __syncthreads();

  // --- Stage 2: LDS transpose  xb[w*TSTR + b] = row b, word w ---------------
  #pragma unroll
  for (int it = 0; it < 16; ++it) {
    int id = it * 128 + t;               // id = b*256 + w
    xb[(id & 255) * TSTR + (id >> 8)] = xlin[id];
  }
  __syncthreads();

  // --- Stage 3: compute 4 consecutive outputs x 8 rows per thread -----------
  const int o0 = obase + t * 4;

  int m[4][6];
  u64 lut[4];
  #pragma unroll
  for (int j = 0; j < 4; ++j) {
    const int* mp = mapping + (size_t)(o0 + j) * 6;
    #pragma unroll
    for (int n = 0; n < 6; ++n) m[j][n] = mp[n];
    lut[j] = lutbits[o0 + j];
  }

  float res[BB][4];
  #pragma unroll
  for (int j = 0; j < 4; ++j) {
    u32 addr[BB] = {0, 0, 0, 0, 0, 0, 0, 0};
    #pragma unroll
    for (int n = 0; n < 6; ++n) {
      int mm = m[j][n];
      int w  = mm >> 5;
      int s  = mm & 31;
      const u32* p = xb + w * TSTR;
      uint4 lo = *(const uint4*)(p);      // word w of rows 0..3
      uint4 hi = *(const uint4*)(p + 4);  // word w of rows 4..7
      addr[0] |= ((lo.x >> s) & 1u) << n;
      addr[1] |= ((lo.y >> s) & 1u) << n;
      addr[2] |= ((lo.z >> s) & 1u) << n;
      addr[3] |= ((lo.w >> s) & 1u) << n;
      addr[4] |= ((hi.x >> s) & 1u) << n;
      addr[5] |= ((hi.y >> s) & 1u) << n;
      addr[6] |= ((hi.z >> s) & 1u) << n;
      addr[7] |= ((hi.w >> s) & 1u) << n;
    }
    #pragma unroll
    for (int b = 0; b < BB; ++b)
      res[b][j] = (float)((lut[j] >> addr[b]) & 1ull);
  }

  #pragma unroll
  for (int b = 0; b < BB; ++b) {
    float4 r = make_float4(res[b][0], res[b][1], res[b][2], res[b][3]);
    *(float4*)(out + (size_t)(bbase + b) * Onum + o0) = r;
  }
}

// -----------------------------------------------------------------------------
// Host launcher. Inputs: x (B,I) f32, mapping (O,6) i32, luts (O,64) f32.
// Output: (B,O) f32. Workspace: xbits (B*I/8 bytes) + lutbits (O*8 bytes).
// -----------------------------------------------------------------------------
extern "C" void kernel_launch(void* const* d_in, const int* in_sizes, int n_in,
                              void* d_out, int out_size, void* d_ws, size_t ws_size,
                              hipStream_t stream) {
  const float* x       = (const float*)d_in[0];
  const int*   mapping = (const int*)d_in[1];
  const float* luts    = (const float*)d_in[2];
  float*       out     = (float*)d_out;

  const int T  = 64;                  // 2^N, N = 6
  const int O  = in_sizes[2] / T;     // 4096
  const int Bn = out_size / O;        // 512
  const int I  = in_sizes[0] / Bn;    // 8192

  u32* xbits   = (u32*)d_ws;                                       // Bn*I/32 words
  u64* lutbits = (u64*)((char*)d_ws + (size_t)Bn * (I / 32) * 4);  // O u64

  (void)n_in; (void)ws_size;

  const int nx = Bn * I;   // 4,194,304
  pack_bits_kernel<<<nx / 256, 256, 0, stream>>>(x, xbits, nx, 0);

  const int nl = O * T;    // 262,144
  pack_bits_kernel<<<nl / 256, 256, 0, stream>>>(luts, (u32*)lutbits, nl, 1);

  dim3 grid(O / OT, Bn / BB);  // (8, 64) = 512 blocks
  lut_forward_kernel<<<grid, 128, 0, stream>>>(xbits, mapping, lutbits, out, O);
}